// GLMAttention_55293408968740
// MI455X (gfx1250) — compile-verified
//
#include <hip/hip_runtime.h>

typedef __attribute__((ext_vector_type(2))) float v2f;
typedef __attribute__((ext_vector_type(8))) float v8f;

#define NHEADS 32
#define NKVH   2
#define HEADD  128
#define BLKSZ  16
#define GQ     16            // NHEADS / NKVH
#define NBPS   256           // blocks per sequence (4096/16)
#define QKVW   4608          // (32 + 2*2) * 128
#define HID    4096
#define NPART  4             // grid-level flash-decode partitions

__device__ __forceinline__ v8f wmma_f32(v2f a, v2f b, v8f c) {
  // D = A(16x4 f32) * B(4x16 f32) + C(16x16 f32)
  return __builtin_amdgcn_wmma_f32_16x16x4_f32(false, a, false, b, (short)0, c,
                                               false, false);
}

// 16B global -> LDS direct copy, tracked by ASYNCcnt (no VGPR round-trip).
// LDS address = low 32 bits of the generic pointer (HW adds wave LDS_BASE).
__device__ __forceinline__ void async_ld16(const float* g, float* l) {
  unsigned lds = (unsigned)(unsigned long long)l;
  asm volatile("global_load_async_to_lds_b128 %0, %1, off"
               :: "v"(lds), "v"(g) : "memory");
}

// ---------------------------------------------------------------------------
// C[32 x N] = A[32 x K] @ B[N x K]^T (+ bias[N]); grid.x = N/16, block = 64.
// Double-buffered async global->LDS staging: 12 b128 issues per wave per
// chunk; wait <=12 keeps the next chunk in flight while computing.
// ---------------------------------------------------------------------------
__global__ __launch_bounds__(64)
void gemm32_nt_wmma(const float* __restrict__ A,
                    const float* __restrict__ B,
                    const float* __restrict__ bias,
                    float* __restrict__ C,
                    int N, int K)
{
  constexpr int KC = 64;
  __shared__ float As[2][32][KC + 4];   // row stride 272 B (16B multiple)
  __shared__ float Bs[2][16][KC + 4];

  const int tid  = threadIdx.x;
  const int wv   = tid >> 5;
  const int lane = tid & 31;
  const int hf   = lane >> 4;
  const int idx  = lane & 15;
  const int n0   = blockIdx.x * 16;
  const int m0   = wv * 16;

  auto issue_chunk = [&](int kc, int bsel) {
    #pragma unroll
    for (int u = 0; u < 8; ++u) {              // A: 32x64 floats = 512 float4
      int i = tid + u * 64;
      int r = i >> 4;
      int c = (i & 15) << 2;
      async_ld16(A + (long)r * K + kc + c, &As[bsel][r][c]);
    }
    #pragma unroll
    for (int u = 0; u < 4; ++u) {              // B: 16x64 floats = 256 float4
      int i = tid + u * 64;
      int r = i >> 4;
      int c = (i & 15) << 2;
      async_ld16(B + (long)(n0 + r) * K + kc + c, &Bs[bsel][r][c]);
    }
  };

  v8f acc = (v8f){};
  int buf = 0;
  issue_chunk(0, 0);
  for (int kc = 0; kc < K; kc += KC, buf ^= 1) {
    if (kc + KC < K) {
      issue_chunk(kc + KC, buf ^ 1);
      asm volatile("s_wait_asynccnt 0xc" ::: "memory");  // current chunk done
    } else {
      asm volatile("s_wait_asynccnt 0x0" ::: "memory");
    }
    __syncthreads();
    #pragma unroll
    for (int kk = 0; kk < KC; kk += 4) {
      v2f a, b;
      a.x = As[buf][m0 + idx][kk + 2 * hf + 0];
      a.y = As[buf][m0 + idx][kk + 2 * hf + 1];
      b.x = Bs[buf][idx][kk + 2 * hf + 0];
      b.y = Bs[buf][idx][kk + 2 * hf + 1];
      acc = wmma_f32(a, b, acc);
    }
    __syncthreads();   // all waves done reading buf before it is re-filled
  }
  float bv = bias ? bias[n0 + idx] : 0.0f;
  #pragma unroll
  for (int r = 0; r < 8; ++r) {
    int m = m0 + r + 8 * hf;
    C[(long)m * N + n0 + idx] = acc[r] + bv;
  }
}

// ---------------------------------------------------------------------------
// Paged GQA flash-decode, partitioned. grid = (NKVH, BATCH, NPART),
// block = 128 (4 waves) -> 16 partitions per (b,g). Each block writes an
// UNNORMALIZED partial (M, L, O) to workspace; attn_combine merges them.
// New token's k/v substituted at slot_mapping (inputs never mutated).
// ---------------------------------------------------------------------------
__global__ __launch_bounds__(128)
void paged_attn_partial(const float* __restrict__ qkv,        // [32 x 4608]
                        const float* __restrict__ k_cache,    // [blk,16,2,128]
                        const float* __restrict__ v_cache,
                        const int*   __restrict__ block_tables, // [32 x 256]
                        const int*   __restrict__ slot_mapping, // [32]
                        const int*   __restrict__ seq_lens,     // [32]
                        float*       __restrict__ part_o,       // [b,g,p,16,128]
                        float*       __restrict__ part_ml)      // [b,g,p,2,16]
{
  __shared__ float kvS[4][16][HEADD + 4];  // per-wave K/V staging (row 528 B)
  __shared__ float pS[4][16][17];          // per-wave P re-layout
  __shared__ float red_m[4][16];
  __shared__ float red_l[4][16];
  __shared__ float outS[16][HEADD];        // cross-wave accumulation

  const int g    = blockIdx.x;
  const int b    = blockIdx.y;
  const int part = blockIdx.z;
  const int tid  = threadIdx.x;
  const int wv   = tid >> 5;
  const int lane = tid & 31;
  const int hf   = lane >> 4;
  const int idx  = lane & 15;

  const int seqlen = seq_lens[b];
  const int ntiles = (seqlen + BLKSZ - 1) / BLKSZ;
  const int slot   = slot_mapping[b];

  // Preload Q fragments: A = Q_group[16 heads x 128]
  v2f qf[32];
  {
    const float* qrow = qkv + (long)b * QKVW + (g * GQ + idx) * HEADD;
    #pragma unroll
    for (int j = 0; j < 32; ++j) {
      qf[j].x = qrow[4 * j + 2 * hf + 0];
      qf[j].y = qrow[4 * j + 2 * hf + 1];
    }
  }
  const float* knew = qkv + (long)b * QKVW + NHEADS * HEADD + g * HEADD;
  const float* vnew = knew + NKVH * HEADD;

  v8f oacc[8];
  #pragma unroll
  for (int d = 0; d < 8; ++d) oacc[d] = (v8f){};
  float m_st[8], l_st[8];
  #pragma unroll
  for (int r = 0; r < 8; ++r) { m_st[r] = -3.0e38f; l_st[r] = 0.0f; }

  const float scale = 0.08838834764831845f;  // 1/sqrt(128)

  for (int t = part * 4 + wv; t < ntiles; t += 4 * NPART) {
    const int p0  = t * BLKSZ;
    const int blk = block_tables[b * NBPS + t];
    const float* kbase = k_cache + (long)blk * (BLKSZ * NKVH * HEADD) + g * HEADD;
    // stage K tile (16 pos x 128) via async global->LDS
    #pragma unroll
    for (int r = 0; r < 16; ++r) {
      int c = lane << 2;
      const float* src = (blk * BLKSZ + r == slot) ? (knew + c)
                                                   : (kbase + r * (NKVH * HEADD) + c);
      async_ld16(src, &kvS[wv][r][c]);
    }
    asm volatile("s_wait_asynccnt 0x0" ::: "memory");  // wave-private buffer

    // S = Q @ K^T  (32 WMMAs over K=128)
    v8f s = (v8f){};
    #pragma unroll
    for (int j = 0; j < 32; ++j) {
      v2f bf;
      bf.x = kvS[wv][idx][4 * j + 2 * hf + 0];
      bf.y = kvS[wv][idx][4 * j + 2 * hf + 1];
      s = wmma_f32(qf[j], bf, s);
    }
    const bool maskme = (p0 + idx) >= seqlen;
    #pragma unroll
    for (int r = 0; r < 8; ++r) s[r] = maskme ? -3.0e38f : s[r] * scale;

    // online softmax (per-row reductions across 16-lane groups)
    float corr[8];
    #pragma unroll
    for (int r = 0; r < 8; ++r) {
      float v = s[r];
      #pragma unroll
      for (int off = 8; off >= 1; off >>= 1) v = fmaxf(v, __shfl_xor(v, off, 16));
      float mnew = fmaxf(m_st[r], v);
      corr[r]   = __expf(m_st[r] - mnew);
      float p   = __expf(s[r] - mnew);
      s[r] = p;
      float rs = p;
      #pragma unroll
      for (int off = 8; off >= 1; off >>= 1) rs += __shfl_xor(rs, off, 16);
      l_st[r] = l_st[r] * corr[r] + rs;
      m_st[r] = mnew;
    }
    #pragma unroll
    for (int d = 0; d < 8; ++d)
      #pragma unroll
      for (int r = 0; r < 8; ++r) oacc[d][r] *= corr[r];

    // re-layout P (D-layout -> A-layout) through LDS
    #pragma unroll
    for (int r = 0; r < 8; ++r) pS[wv][r + 8 * hf][idx] = s[r];

    // stage V tile (safe: all K-phase ds reads are consumed by the WMMAs)
    const float* vbase = v_cache + (long)blk * (BLKSZ * NKVH * HEADD) + g * HEADD;
    #pragma unroll
    for (int r = 0; r < 16; ++r) {
      int c = lane << 2;
      const float* src = (blk * BLKSZ + r == slot) ? (vnew + c)
                                                   : (vbase + r * (NKVH * HEADD) + c);
      async_ld16(src, &kvS[wv][r][c]);
    }
    asm volatile("s_wait_asynccnt 0x0" ::: "memory");

    // O += P @ V  (8 d-tiles x 4 K-steps)
    #pragma unroll
    for (int dt = 0; dt < 8; ++dt) {
      #pragma unroll
      for (int ks = 0; ks < 16; ks += 4) {
        v2f a, bf;
        a.x  = pS[wv][idx][ks + 2 * hf + 0];
        a.y  = pS[wv][idx][ks + 2 * hf + 1];
        bf.x = kvS[wv][ks + 2 * hf + 0][dt * 16 + idx];
        bf.y = kvS[wv][ks + 2 * hf + 1][dt * 16 + idx];
        oacc[dt] = wmma_f32(a, bf, oacc[dt]);
      }
    }
  }

  // merge 4 in-block waves -> unnormalized partial (M_p, L_p, O_p)
  for (int i = tid; i < 16 * HEADD; i += 128) outS[i >> 7][i & 127] = 0.0f;
  if (idx == 0) {
    #pragma unroll
    for (int r = 0; r < 8; ++r) {
      red_m[wv][r + 8 * hf] = m_st[r];
      red_l[wv][r + 8 * hf] = l_st[r];
    }
  }
  __syncthreads();
  float fac[8];
  #pragma unroll
  for (int r = 0; r < 8; ++r) {
    int m = r + 8 * hf;
    float Mp = red_m[0][m];
    #pragma unroll
    for (int w = 1; w < 4; ++w) Mp = fmaxf(Mp, red_m[w][m]);
    fac[r] = __expf(m_st[r] - Mp);   // no normalization here
  }
  #pragma unroll
  for (int dt = 0; dt < 8; ++dt)
    #pragma unroll
    for (int r = 0; r < 8; ++r)
      atomicAdd(&outS[r + 8 * hf][dt * 16 + idx], oacc[dt][r] * fac[r]);
  __syncthreads();

  const long pbase = (long)((b * NKVH + g) * NPART + part);
  if (tid < 16) {
    const int m = tid;
    float Mp = red_m[0][m];
    #pragma unroll
    for (int w = 1; w < 4; ++w) Mp = fmaxf(Mp, red_m[w][m]);
    float Lp = 0.0f;
    #pragma unroll
    for (int w = 0; w < 4; ++w) Lp += red_l[w][m] * __expf(red_m[w][m] - Mp);
    part_ml[pbase * 32 + m]      = Mp;
    part_ml[pbase * 32 + 16 + m] = Lp;
  }
  for (int i = tid; i < 16 * HEADD; i += 128)
    part_o[pbase * (16 * HEADD) + i] = outS[i >> 7][i & 127];
}

// ---------------------------------------------------------------------------
// Merge NPART partials: out = sum_p e^{M_p-M} O_p / sum_p e^{M_p-M} L_p.
// grid = (NKVH, BATCH), block = 128.
// ---------------------------------------------------------------------------
__global__ __launch_bounds__(128)
void attn_combine(const float* __restrict__ part_o,
                  const float* __restrict__ part_ml,
                  float* __restrict__ attn_out)
{
  const int g = blockIdx.x, b = blockIdx.y, tid = threadIdx.x;
  const long base = (long)(b * NKVH + g) * NPART;
  for (int i = tid; i < 16 * HEADD; i += 128) {
    const int m = i >> 7;
    float M = -3.0e38f;
    #pragma unroll
    for (int p = 0; p < NPART; ++p) M = fmaxf(M, part_ml[(base + p) * 32 + m]);
    float L = 0.0f, num = 0.0f;
    #pragma unroll
    for (int p = 0; p < NPART; ++p) {
      const float w = __expf(part_ml[(base + p) * 32 + m] - M);
      L   += w * part_ml[(base + p) * 32 + 16 + m];
      num += w * part_o[(base + p) * (16 * HEADD) + i];
    }
    attn_out[(long)b * (NHEADS * HEADD) + g * (GQ * HEADD) + i] = num / L;
  }
}

// ---------------------------------------------------------------------------
extern "C" void kernel_launch(void* const* d_in, const int* in_sizes, int n_in,
                              void* d_out, int out_size, void* d_ws, size_t ws_size,
                              hipStream_t stream) {
  const float* x            = (const float*)d_in[0];
  const float* w_qkv        = (const float*)d_in[1];
  const float* b_qkv        = (const float*)d_in[2];
  const float* w_dense      = (const float*)d_in[3];
  const float* k_cache      = (const float*)d_in[4];
  const float* v_cache      = (const float*)d_in[5];
  const int*   slot_mapping = (const int*)d_in[6];
  const int*   block_tables = (const int*)d_in[7];
  const int*   seq_lens     = (const int*)d_in[8];
  float*       out          = (float*)d_out;

  float* qkv     = (float*)d_ws;                       // 32*4608
  float* attn    = qkv + 32 * QKVW;                    // 32*4096
  float* part_o  = attn + 32 * HID;                    // 32*2*NPART*2048
  float* part_ml = part_o + 32 * NKVH * NPART * 2048;  // 32*2*NPART*32

  // 1) QKV projection (+bias)
  gemm32_nt_wmma<<<dim3(QKVW / 16), 64, 0, stream>>>(x, w_qkv, b_qkv, qkv,
                                                     QKVW, HID);
  // 2) partitioned paged flash-decode attention
  paged_attn_partial<<<dim3(NKVH, 32, NPART), 128, 0, stream>>>(
      qkv, k_cache, v_cache, block_tables, slot_mapping, seq_lens,
      part_o, part_ml);
  attn_combine<<<dim3(NKVH, 32), 128, 0, stream>>>(part_o, part_ml, attn);
  // 3) output projection
  gemm32_nt_wmma<<<dim3(HID / 16), 64, 0, stream>>>(attn, w_dense, nullptr, out,
                                                    HID, HID);
}